// STDP_79809082294373
// MI455X (gfx1250) — compile-verified
//
#include <hip/hip_runtime.h>

// dw = alpha * pre^T * post  (single GEMM; both STDP einsums are scalar
// multiples of the same Gram matrix).
// M = S = 8192, N = T = 8192, K = B = 256, alpha = 7.8125e-9.

typedef float v2f __attribute__((ext_vector_type(2)));
typedef float v8f __attribute__((ext_vector_type(8)));

#define KDIM 256
#define SDIM 8192
#define TDIM 8192
#define ALPHA 7.8125e-9f

#define KC 16                      // K-chunk staged per LDS buffer
#define NCHUNK (KDIM / KC)         // 16

// Padded LDS row strides (floats): +8 pad makes a 2-row offset = 16 banks,
// so lane halves (rows r and r+2) of a WMMA fragment read hit disjoint banks.
#define A_STRIDE 136               // 128 cols + 8 pad
#define B_STRIDE 264               // 256 cols + 8 pad
#define A_FLOATS (KC * A_STRIDE)   // 2176 floats = 8704 B
#define B_FLOATS (KC * B_STRIDE)   // 4224 floats = 16896 B
#define BUF_FLOATS (A_FLOATS + B_FLOATS)  // 6400 floats = 25600 B per buffer

// One 16-byte async DMA copy: global -> LDS, tracked by ASYNCcnt.
__device__ __forceinline__ void async_cp16(unsigned lds_byte_off, const float* g) {
    asm volatile("global_load_async_to_lds_b128 %0, %1, off"
                 :: "v"(lds_byte_off), "v"((unsigned long long)(size_t)g)
                 : "memory");
}

__device__ __forceinline__ void wait_async0() {
    asm volatile("s_wait_asynccnt 0x0" ::: "memory");
}

// Stage one K-chunk (A: 16x128, B: 16x256) into LDS; 6 b128 copies/thread.
__device__ __forceinline__ void stage_chunk(const float* __restrict__ A,
                                            const float* __restrict__ P,
                                            int m0blk, int n0blk, int k0,
                                            unsigned lds_base_bytes, int tid) {
    #pragma unroll
    for (int r = 0; r < 2; ++r) {            // 512 transfers for A
        const int i   = tid + r * 256;
        const int row = i >> 5;
        const int c16 = i & 31;
        async_cp16(lds_base_bytes + row * (A_STRIDE * 4) + c16 * 16,
                   A + (size_t)(k0 + row) * SDIM + m0blk + c16 * 4);
    }
    #pragma unroll
    for (int r = 0; r < 4; ++r) {            // 1024 transfers for B
        const int i   = tid + r * 256;
        const int row = i >> 6;
        const int c64 = i & 63;
        async_cp16(lds_base_bytes + A_FLOATS * 4 + row * (B_STRIDE * 4) + c64 * 16,
                   P + (size_t)(k0 + row) * TDIM + n0blk + c64 * 4);
    }
}

// 8 waves (wave32) as 2(M) x 4(N); wave tile 64x64 -> block tile 128x256.
__global__ __launch_bounds__(256)
void stdp_gemm_wmma_async(const float* __restrict__ A,   // (K, S) pre
                          const float* __restrict__ P,   // (K, T) post
                          float* __restrict__ C)         // (S, T) dw
{
    __shared__ float smem[2 * BUF_FLOATS];               // 51200 B, double-buffered

    const int tid  = threadIdx.x;
    const int wave = tid >> 5;
    const int lane = tid & 31;
    const int half = lane >> 4;
    const int l    = lane & 15;
    const int wm   = wave >> 2;      // 0..1
    const int wn   = wave & 3;       // 0..3

    const int m0blk = blockIdx.y * 128;
    const int n0blk = blockIdx.x * 256;

    // Low 32 bits of the flat address of an LDS object are the LDS byte offset.
    const unsigned lds0 = (unsigned)(size_t)(void*)&smem[0];

    v8f acc[4][4];
    #pragma unroll
    for (int i = 0; i < 4; ++i)
        #pragma unroll
        for (int j = 0; j < 4; ++j)
            acc[i][j] = {};

    stage_chunk(A, P, m0blk, n0blk, 0, lds0, tid);

    for (int c = 0; c < NCHUNK; ++c) {
        wait_async0();        // my chunk-c copies have landed in LDS
        __syncthreads();      // everyone's landed; everyone done reading buf (c-1)

        if (c + 1 < NCHUNK)
            stage_chunk(A, P, m0blk, n0blk, (c + 1) * KC,
                        lds0 + (unsigned)(((c + 1) & 1) * (BUF_FLOATS * 4)), tid);

        const float* __restrict__ Ab = &smem[(c & 1) * BUF_FLOATS];
        const float* __restrict__ Bb = Ab + A_FLOATS;

        #pragma unroll
        for (int kk = 0; kk < KC / 4; ++kk) {
            const int r0 = kk * 4 + 2 * half;   // K rows for this lane half

            v2f af[4];
            #pragma unroll
            for (int mi = 0; mi < 4; ++mi) {
                const int m = wm * 64 + mi * 16 + l;
                af[mi].x = Ab[r0 * A_STRIDE + m];
                af[mi].y = Ab[(r0 + 1) * A_STRIDE + m];
            }
            v2f bf[4];
            #pragma unroll
            for (int ni = 0; ni < 4; ++ni) {
                const int n = wn * 64 + ni * 16 + l;
                bf[ni].x = Bb[r0 * B_STRIDE + n];
                bf[ni].y = Bb[(r0 + 1) * B_STRIDE + n];
            }

            #pragma unroll
            for (int mi = 0; mi < 4; ++mi)
                #pragma unroll
                for (int ni = 0; ni < 4; ++ni)
                    acc[mi][ni] = __builtin_amdgcn_wmma_f32_16x16x4_f32(
                        false, af[mi], false, bf[ni],
                        (short)0, acc[mi][ni], false, false);
        }
    }

    // C/D 16x16 f32 layout: VGPR v -> row (v + 8*half), lane l -> col.
    #pragma unroll
    for (int mi = 0; mi < 4; ++mi) {
        #pragma unroll
        for (int ni = 0; ni < 4; ++ni) {
            const int col = n0blk + wn * 64 + ni * 16 + l;
            #pragma unroll
            for (int v = 0; v < 8; ++v) {
                const int row = m0blk + wm * 64 + mi * 16 + half * 8 + v;
                C[(size_t)row * TDIM + col] = acc[mi][ni][v] * ALPHA;
            }
        }
    }
}

extern "C" void kernel_launch(void* const* d_in, const int* in_sizes, int n_in,
                              void* d_out, int out_size, void* d_ws, size_t ws_size,
                              hipStream_t stream) {
    // inputs: neurotransmitters (B,S), neuromodulators [unused],
    //         action_potential (B,T), synaptic_weights [shape only]
    const float* pre  = (const float*)d_in[0];
    const float* post = (const float*)d_in[2];
    float* out = (float*)d_out;

    dim3 grid(TDIM / 256, SDIM / 128);
    stdp_gemm_wmma_async<<<grid, dim3(256), 0, stream>>>(pre, post, out);
}